// LDMMolDesignRAGfull_77008763617336
// MI455X (gfx1250) — compile-verified
//
#include <hip/hip_runtime.h>
#include <hip/hip_bf16.h>

#define N_ATOMS  500000
#define N_EDGES  1000000
#define N_BLOCKS 100000
#define HIDDEN   128

typedef float v2f __attribute__((ext_vector_type(2)));
typedef float v8f __attribute__((ext_vector_type(8)));

// ---------------------------------------------------------------- utilities
__global__ void zero_f32_kernel(float* __restrict__ p, int n) {
    int i = blockIdx.x * blockDim.x + threadIdx.x;
    if (i < n) p[i] = 0.0f;
}

// ------------------------------------------------------------- init kernel
// hPre[a][:] = (1+eps) * atom_table[A[a]][:]   (the "(1+eps)*H" term)
// ctx[a]     = !generate_mask[block_ids[a]]
// cnt[bid]  += 1
// One wave (32 lanes) per atom; each lane handles 4 consecutive features.
__global__ void init_kernel(const int* __restrict__ A,
                            const int* __restrict__ block_ids,
                            const int* __restrict__ gmask,
                            const float* __restrict__ atom_table,
                            const float* __restrict__ epsp,
                            float* __restrict__ hPre,
                            int* __restrict__ ctx,
                            float* __restrict__ cnt)
{
    int t    = blockIdx.x * blockDim.x + threadIdx.x;
    int atom = t >> 5;
    int lane = t & 31;
    if (atom >= N_ATOMS) return;

    float scale = 1.0f + epsp[0];
    int vocab   = A[atom];
    float4 x = ((const float4*)(atom_table + (long)vocab * HIDDEN))[lane];
    x.x *= scale; x.y *= scale; x.z *= scale; x.w *= scale;
    ((float4*)(hPre + (long)atom * HIDDEN))[lane] = x;

    if (lane == 0) {
        int bid = block_ids[atom];
        ctx[atom] = gmask[bid] ? 0 : 1;
        atomicAdd(&cnt[bid], 1.0f);
    }
}

// ------------------------------------------------------------- edge kernel
// msg = relu(atom_table[A[src]] + bond_table[btype]); scatter-add into hPre[dst].
// One wave per edge, float4 per lane. atom_table (64KB) and bond_table (2.5KB)
// are L2-resident, so regathering beats materializing a 256MB H array.
__global__ void edge_kernel(const int* __restrict__ bonds,
                            const int* __restrict__ A,
                            const int* __restrict__ ctx,
                            const float* __restrict__ atom_table,
                            const float* __restrict__ bond_table,
                            float* __restrict__ hPre)
{
    int t    = blockIdx.x * blockDim.x + threadIdx.x;
    int e    = t >> 5;
    int lane = t & 31;
    if (e >= N_EDGES) return;

    int src = bonds[e * 3 + 0];
    int dst = bonds[e * 3 + 1];
    int bt  = bonds[e * 3 + 2];
    if (!(ctx[src] & ctx[dst])) return;   // edge fully masked: skip (uniform branch)

    int vocab = A[src];
    float4 h  = ((const float4*)(atom_table + (long)vocab * HIDDEN))[lane];
    float4 eb = ((const float4*)(bond_table + (long)bt    * HIDDEN))[lane];
    float4 m;
    m.x = fmaxf(h.x + eb.x, 0.0f);
    m.y = fmaxf(h.y + eb.y, 0.0f);
    m.z = fmaxf(h.z + eb.z, 0.0f);
    m.w = fmaxf(h.w + eb.w, 0.0f);

    float* out = hPre + (long)dst * HIDDEN + lane * 4;
    atomicAdd(out + 0, m.x);
    atomicAdd(out + 1, m.y);
    atomicAdd(out + 2, m.z);
    atomicAdd(out + 3, m.w);
}

// --------------------------------------------------------- fused MLP kernel
// Per 128-row tile:  T = relu(h @ W1 + b1);  h2 = T @ W2 + b2;
// then scatter h2 rows into blkSum[block_ids[a]] with atomics (h2 never hits HBM).
// 256 threads = 8 waves; wave w owns M-rows [16w,16w+16); loops 8 N-tiles of 16.
// FP32 WMMA 16x16x4, K-loop of 32.
// Dynamic LDS: sA(64KB) | sW(64KB, transposed weights) | sT(64KB)  = 192KB/WGP.
__global__ __launch_bounds__(256) void mlp_block_kernel(
        const float* __restrict__ hPre,
        const float* __restrict__ W1, const float* __restrict__ b1,
        const float* __restrict__ W2, const float* __restrict__ b2,
        const int*  __restrict__ block_ids,
        float* __restrict__ blkSum)
{
    extern __shared__ float smem[];
    float* sA = smem;                       // [128][128]  input tile
    float* sW = smem + 128 * HIDDEN;        // [128][128]  W^T (row n, col k)
    float* sT = smem + 2 * 128 * HIDDEN;    // [128][128]  relu intermediate

    const int base = blockIdx.x * 128;

    // ---- stage input tile (zero-pad ragged tail rows)
    for (int i = threadIdx.x; i < 128 * HIDDEN / 4; i += 256) {
        int a = base + (i >> 5);            // 32 float4 per row
        float4 v4 = make_float4(0.f, 0.f, 0.f, 0.f);
        if (a < N_ATOMS) v4 = ((const float4*)hPre)[(long)a * (HIDDEN / 4) + (i & 31)];
        ((float4*)sA)[i] = v4;
    }
    // ---- stage W1 transposed: sW[n*128 + k] = W1[k*128 + n]
    for (int i = threadIdx.x; i < HIDDEN * HIDDEN; i += 256) {
        int k = i >> 7, n = i & 127;
        sW[n * HIDDEN + k] = W1[i];
    }
    __syncthreads();

    const int wave = threadIdx.x >> 5;
    const int lane = threadIdx.x & 31;
    const int j    = lane & 15;             // N index (B/C/D striping), M index (A striping)
    const int half = lane >> 4;             // hi/lo 16-lane group
    const int mBase = wave * 16;
    const int mRow  = mBase + j;            // A-fragment source row

    // =============================== GEMM 1: T = relu(h @ W1 + b1)
    for (int nt = 0; nt < 8; ++nt) {
        const int nBase = nt << 4;
        const float bias = b1[nBase + j];
        v8f c = { bias, bias, bias, bias, bias, bias, bias, bias };
        #pragma unroll 8
        for (int kk = 0; kk < HIDDEN / 4; ++kk) {
            const int k0 = (kk << 2) + (half << 1);
            // A 16x4: lane j holds row M=mRow, VGPR v = K = k0+v  (ISA 7.12.2)
            v2f a = { sA[mRow * HIDDEN + k0], sA[mRow * HIDDEN + k0 + 1] };
            // B 4x16: lane j holds col N=nBase+j, VGPR v = K = k0+v (W stored transposed)
            v2f b = { sW[(nBase + j) * HIDDEN + k0], sW[(nBase + j) * HIDDEN + k0 + 1] };
            c = __builtin_amdgcn_wmma_f32_16x16x4_f32(false, a, false, b,
                                                      (short)0, c, false, false);
        }
        // C/D 16x16 f32: VGPR v, this lane -> (M = v + half*8, N = j)
        #pragma unroll
        for (int v = 0; v < 8; ++v) {
            int m = mBase + half * 8 + v;
            sT[m * HIDDEN + nBase + j] = fmaxf(c[v], 0.0f);
        }
    }
    __syncthreads();

    // ---- stage W2 transposed (reuse sW)
    for (int i = threadIdx.x; i < HIDDEN * HIDDEN; i += 256) {
        int k = i >> 7, n = i & 127;
        sW[n * HIDDEN + k] = W2[i];
    }
    __syncthreads();

    // =============================== GEMM 2: h2 = T @ W2 + b2, scatter to blocks
    for (int nt = 0; nt < 8; ++nt) {
        const int nBase = nt << 4;
        const float bias = b2[nBase + j];
        v8f c = { bias, bias, bias, bias, bias, bias, bias, bias };
        #pragma unroll 8
        for (int kk = 0; kk < HIDDEN / 4; ++kk) {
            const int k0 = (kk << 2) + (half << 1);
            v2f a = { sT[mRow * HIDDEN + k0], sT[mRow * HIDDEN + k0 + 1] };
            v2f b = { sW[(nBase + j) * HIDDEN + k0], sW[(nBase + j) * HIDDEN + k0 + 1] };
            c = __builtin_amdgcn_wmma_f32_16x16x4_f32(false, a, false, b,
                                                      (short)0, c, false, false);
        }
        #pragma unroll
        for (int v = 0; v < 8; ++v) {
            int m    = mBase + half * 8 + v;
            int atom = base + m;
            if (atom < N_ATOMS) {
                int bid = block_ids[atom];
                atomicAdd(&blkSum[(long)bid * HIDDEN + nBase + j], c[v]);
            }
        }
    }
}

// --------------------------------------------------------- finalize kernel
// out = generate_mask ? 0 : blkSum / sqrt(max(cnt,1))
__global__ void finalize_kernel(const int* __restrict__ gmask,
                                const float* __restrict__ cnt,
                                float* __restrict__ out)
{
    int i = blockIdx.x * blockDim.x + threadIdx.x;
    if (i >= N_BLOCKS * HIDDEN) return;
    int b = i >> 7;
    float v = 0.0f;
    if (!gmask[b]) {
        float c = fmaxf(cnt[b], 1.0f);
        v = out[i] * rsqrtf(c);
    }
    out[i] = v;
}

// ------------------------------------------------------------------ launch
extern "C" void kernel_launch(void* const* d_in, const int* in_sizes, int n_in,
                              void* d_out, int out_size, void* d_ws, size_t ws_size,
                              hipStream_t stream) {
    // setup_inputs() order:
    const int*   A          = (const int*)  d_in[0];   // [N_ATOMS]
    const int*   bonds      = (const int*)  d_in[1];   // [N_EDGES][3]
    const int*   block_ids  = (const int*)  d_in[2];   // [N_ATOMS] (sorted)
    const int*   gmask      = (const int*)  d_in[3];   // [N_BLOCKS] bool
    const float* atom_table = (const float*)d_in[4];   // [128][128]
    const float* bond_table = (const float*)d_in[5];   // [5][128]
    const float* eps        = (const float*)d_in[6];   // scalar
    const float* W1         = (const float*)d_in[7];   // [128][128]
    const float* b1         = (const float*)d_in[8];   // [128]
    const float* W2         = (const float*)d_in[9];   // [128][128]
    const float* b2         = (const float*)d_in[10];  // [128]
    float*       out        = (float*)d_out;           // [N_BLOCKS][128]

    // workspace layout (16B aligned): hPre | ctx | cnt
    float* hPre = (float*)d_ws;                                      // 256,000,000 B
    int*   ctx  = (int*)  ((char*)d_ws + (size_t)N_ATOMS * HIDDEN * 4);   // 2,000,000 B
    float* cnt  = (float*)((char*)d_ws + (size_t)N_ATOMS * HIDDEN * 4
                                       + (size_t)N_ATOMS * 4);            //   400,000 B

    // 0) zero accumulators (harness poisons buffers with 0xAA)
    {
        int n = N_BLOCKS * HIDDEN;
        zero_f32_kernel<<<(n + 255) / 256, 256, 0, stream>>>(out, n);
        zero_f32_kernel<<<(N_BLOCKS + 255) / 256, 256, 0, stream>>>(cnt, N_BLOCKS);
    }
    // 1) hPre = (1+eps)*H, ctx flags, block counts
    {
        long threads = (long)N_ATOMS * 32;
        init_kernel<<<(int)((threads + 255) / 256), 256, 0, stream>>>(
            A, block_ids, gmask, atom_table, eps, hPre, ctx, cnt);
    }
    // 2) edge messages -> scatter into hPre
    {
        long threads = (long)N_EDGES * 32;
        edge_kernel<<<(int)((threads + 255) / 256), 256, 0, stream>>>(
            bonds, A, ctx, atom_table, bond_table, hPre);
    }
    // 3) fused MLP (two FP32 WMMA GEMMs) + block segment-sum
    {
        int blocks = (N_ATOMS + 127) / 128;
        size_t lds = (size_t)3 * 128 * HIDDEN * sizeof(float);  // 192 KB
        mlp_block_kernel<<<blocks, 256, lds, stream>>>(
            hPre, W1, b1, W2, b2, block_ids, out);
    }
    // 4) normalize + mask
    {
        int n = N_BLOCKS * HIDDEN;
        finalize_kernel<<<(n + 255) / 256, 256, 0, stream>>>(gmask, cnt, out);
    }
}